// CGNN_more_fc_70214125355154
// MI455X (gfx1250) — compile-verified
//
#include <hip/hip_runtime.h>

typedef __attribute__((ext_vector_type(2))) float v2f;
typedef __attribute__((ext_vector_type(8))) float v8f;

#define NN 50000
#define NE 800000

// ---------------- degree / norm ----------------
__global__ void init_deg(float* deg, int n) {
  int i = blockIdx.x * blockDim.x + threadIdx.x;
  if (i < n) deg[i] = 1.0f;                       // self-loop
}

__global__ void count_deg(const long long* __restrict__ ei, float* deg, int nE) {
  int e = blockIdx.x * blockDim.x + threadIdx.x;
  if (e < nE) {
    int dst = (int)ei[(size_t)nE + e];
    unsafeAtomicAdd(deg + dst, 1.0f);
  }
}

__global__ void rsqrt_deg(float* deg, int n) {
  int i = blockIdx.x * blockDim.x + threadIdx.x;
  if (i < n) deg[i] = rsqrtf(deg[i]);
}

// ---------------- aggregation (feature dim fixed at 128) ----------------
// agg[i,f] = h[i,f] * dinv[i]^2   (self-loop term, also zero-initializes)
__global__ void self_loop_init(const float* __restrict__ h, const float* __restrict__ dinv,
                               float* __restrict__ agg, int n) {
  int idx = blockIdx.x * blockDim.x + threadIdx.x;
  if (idx < n * 128) {
    int i = idx >> 7;
    float d = dinv[i];
    agg[idx] = h[idx] * d * d;
  }
}

// agg[dst,f] += h[src,f] * dinv[src]*dinv[dst]  (128 lanes per edge, coalesced)
__global__ void edge_scatter(const long long* __restrict__ ei, const float* __restrict__ dinv,
                             const float* __restrict__ h, float* __restrict__ agg, int nE) {
  int t = blockIdx.x * blockDim.x + threadIdx.x;
  int e = t >> 7;
  int f = t & 127;
  if (e < nE) {
    int src = (int)ei[e];
    int dst = (int)ei[(size_t)nE + e];
    float norm = dinv[src] * dinv[dst];
    unsafeAtomicAdd(agg + (size_t)dst * 128 + f, h[(size_t)src * 128 + f] * norm);
  }
}

// ---------------- WMMA f32 GEMM: C[M,N] = A[M,K] @ B[K,N] (+bias, leaky) ----------------
// block = 256 threads = 8 waves; wave w -> rows [blockIdx.y*128 + w*16, +16), cols [blockIdx.x*64, +64)
template <bool BIAS_LEAKY>
__global__ __launch_bounds__(256) void wmma_gemm(const float* __restrict__ A,
                                                 const float* __restrict__ B,
                                                 const float* __restrict__ bias,
                                                 float* __restrict__ C,
                                                 int M, int N, int K) {
  int lane  = threadIdx.x & 31;
  int wave  = threadIdx.x >> 5;
  int mBase = blockIdx.y * 128 + wave * 16;
  int nBase = blockIdx.x * 64;
  if (mBase >= M) return;                          // M % 16 == 0 so full tiles only

  int l15   = lane & 15;
  int kHalf = (lane >> 4) << 1;                    // 0 or 2

  v8f acc0 = {}, acc1 = {}, acc2 = {}, acc3 = {};

  const float* Arow = A + (size_t)(mBase + l15) * K;
  for (int k = 0; k < K; k += 4) {
    int kA = k + kHalf;
    // A frag: lane holds A[m][kA], A[m][kA+1] (8B contiguous)
    v2f a = *(const v2f*)(Arow + kA);
    // B frags: lane holds B[kA][n], B[kA+1][n]
    const float* Bp = B + (size_t)kA * N + nBase + l15;
    v2f b0 = { Bp[0],  Bp[0 + N]  };
    v2f b1 = { Bp[16], Bp[16 + N] };
    v2f b2 = { Bp[32], Bp[32 + N] };
    v2f b3 = { Bp[48], Bp[48 + N] };
    acc0 = __builtin_amdgcn_wmma_f32_16x16x4_f32(false, a, false, b0, (short)0, acc0, false, false);
    acc1 = __builtin_amdgcn_wmma_f32_16x16x4_f32(false, a, false, b1, (short)0, acc1, false, false);
    acc2 = __builtin_amdgcn_wmma_f32_16x16x4_f32(false, a, false, b2, (short)0, acc2, false, false);
    acc3 = __builtin_amdgcn_wmma_f32_16x16x4_f32(false, a, false, b3, (short)0, acc3, false, false);
  }

  // C/D layout: element v of lane -> row mBase + v + 8*(lane>>4), col nBase + nt*16 + (lane&15)
  int rBase = mBase + ((lane >> 4) << 3);
#pragma unroll
  for (int nt = 0; nt < 4; ++nt) {
    v8f c = (nt == 0) ? acc0 : (nt == 1) ? acc1 : (nt == 2) ? acc2 : acc3;
    int col = nBase + nt * 16 + l15;
    float bv = BIAS_LEAKY ? bias[col] : 0.0f;
#pragma unroll
    for (int v = 0; v < 8; ++v) {
      float val = c[v];
      if (BIAS_LEAKY) {
        val += bv;
        val = (val >= 0.0f) ? val : 0.01f * val;
      }
      C[(size_t)(rBase + v) * N + col] = val;
    }
  }
}

// ---------------- elementwise bias + leaky (feature dim 128) ----------------
__global__ void bias_leaky128(float* __restrict__ h, const float* __restrict__ bias, int n) {
  int idx = blockIdx.x * blockDim.x + threadIdx.x;
  if (idx < n * 128) {
    float v = h[idx] + bias[idx & 127];
    h[idx] = (v >= 0.0f) ? v : 0.01f * v;
  }
}

// ---------------- transpose fc1_w [256,128] -> [128,256] ----------------
__global__ void transpose_fc1(const float* __restrict__ w, float* __restrict__ wt) {
  int idx = blockIdx.x * blockDim.x + threadIdx.x;
  if (idx < 256 * 128) {
    int n = idx >> 7, k = idx & 127;
    wt[k * 256 + n] = w[idx];
  }
}

// ---------------- final: out[i] = dot(h3[i,:256], fc2_w) + fc2_b ----------------
__global__ void final_dot(const float* __restrict__ h3, const float* __restrict__ w2,
                          const float* __restrict__ b2, float* __restrict__ out, int n) {
  int lane = threadIdx.x & 31;
  int node = (blockIdx.x * blockDim.x + threadIdx.x) >> 5;
  if (node >= n) return;
  const float* row = h3 + (size_t)node * 256;
  float s = 0.0f;
#pragma unroll
  for (int j = 0; j < 8; ++j) {
    int idx = lane + j * 32;
    s += row[idx] * w2[idx];
  }
#pragma unroll
  for (int off = 16; off > 0; off >>= 1) s += __shfl_down(s, off, 32);
  if (lane == 0) out[node] = s + b2[0];
}

extern "C" void kernel_launch(void* const* d_in, const int* in_sizes, int n_in,
                              void* d_out, int out_size, void* d_ws, size_t ws_size,
                              hipStream_t stream) {
  const float*     x    = (const float*)d_in[0];
  const long long* ei   = (const long long*)d_in[1];
  const float*     W1   = (const float*)d_in[2];
  const float*     b1   = (const float*)d_in[3];
  const float*     W3   = (const float*)d_in[4];
  const float*     b3   = (const float*)d_in[5];
  const float*     fc1w = (const float*)d_in[6];
  const float*     fc1b = (const float*)d_in[7];
  const float*     fc2w = (const float*)d_in[8];
  const float*     fc2b = (const float*)d_in[9];
  float* out = (float*)d_out;

  const int N = NN, E = NE;
  float* ws   = (float*)d_ws;
  float* dinv = ws;                             // N
  float* fc1T = dinv + N;                       // 128*256
  float* bufA = fc1T + 128 * 256;               // N*512  (h1, later h3[N,256])
  float* bufB = bufA + (size_t)N * 512;         // N*128  (agg1, later agg2/h2)
  float* bufC = bufB + (size_t)N * 128;         // N*128  (t2)

  // normalization: deg = 1 + in-degree(dst); dinv = rsqrt(deg)
  init_deg<<<(N + 255) / 256, 256, 0, stream>>>(dinv, N);
  count_deg<<<(E + 255) / 256, 256, 0, stream>>>(ei, dinv, E);
  rsqrt_deg<<<(N + 255) / 256, 256, 0, stream>>>(dinv, N);
  transpose_fc1<<<(256 * 128 + 255) / 256, 256, 0, stream>>>(fc1w, fc1T);

  // layer 1: aggregate x (128-dim) then GEMM -> h1 = leaky(agg1 @ W1 + b1)
  self_loop_init<<<(N * 128 + 255) / 256, 256, 0, stream>>>(x, dinv, bufB, N);
  edge_scatter<<<(E * 128) / 256, 256, 0, stream>>>(ei, dinv, x, bufB, E);
  wmma_gemm<true><<<dim3(512 / 64, (N + 127) / 128), 256, 0, stream>>>(bufB, W1, b1, bufA, N, 512, 128);

  // layer 2: GEMM first (512->128), then aggregate in 128-dim
  wmma_gemm<false><<<dim3(128 / 64, (N + 127) / 128), 256, 0, stream>>>(bufA, W3, nullptr, bufC, N, 128, 512);
  self_loop_init<<<(N * 128 + 255) / 256, 256, 0, stream>>>(bufC, dinv, bufB, N);
  edge_scatter<<<(E * 128) / 256, 256, 0, stream>>>(ei, dinv, bufC, bufB, E);
  bias_leaky128<<<(N * 128 + 255) / 256, 256, 0, stream>>>(bufB, b3, N);

  // fc1: h3 = leaky(h2 @ fc1T + fc1b)  [N,256]
  wmma_gemm<true><<<dim3(256 / 64, (N + 127) / 128), 256, 0, stream>>>(bufB, fc1T, fc1b, bufA, N, 256, 128);

  // fc2: out[i] = dot(h3[i], fc2w) + fc2b
  final_dot<<<(N * 32 + 255) / 256, 256, 0, stream>>>(bufA, fc2w, fc2b, out, N);
}